// DSE1_31739808318045
// MI455X (gfx1250) — compile-verified
//
#include <hip/hip_runtime.h>
#include <math.h>

// ---------------------------------------------------------------------------
// DiffPool-style GNN forward for MI455X (gfx1250, wave32).
//  - Dense GEMMs (h[n x 128] @ W + b) use V_WMMA_F32_16X16X4_F32 (f32-exact).
//    B fragments are held in VGPRs across row tiles; A tiles staged in LDS.
//  - Sparse segment ops use f32 global atomics (resolve in 192MB L2) with
//    global_prefetch_b8 ahead-of-gather on the random edge list.
//  - Coarse-level math collapsed exactly (z2 rows are a broadcast mean).
// ---------------------------------------------------------------------------

typedef __attribute__((ext_vector_type(2))) float v2f;
typedef __attribute__((ext_vector_type(8))) float v8f;

#define C1 256
#define C2 32
#define NEG_SLOPE 0.01f
#define EPS_NORM 1e-12f
#define ROW_TILES_PER_BLOCK 4
#define PF_DIST 1024   // edge-gather prefetch distance

// ---------------------------------------------------------------------------
// WMMA GEMM: out[n x Cout] = h[n x 128] @ W[128 x Cout] + b
// NT = column tiles per wave (Cout = NT*8*16). block = 256 (8 waves).
// Each block processes ROW_TILES_PER_BLOCK row tiles of 16; B fragments for
// the wave's column tiles are preloaded into registers once and reused.
// ---------------------------------------------------------------------------
template <int NT>
__global__ __launch_bounds__(256)
void gnn_wmma_gemm_bias(const float* __restrict__ h, const float* __restrict__ W,
                        const float* __restrict__ b, float* __restrict__ out,
                        int n, int Cout) {
  __shared__ float hs[16][130];           // padded: stride 130 -> no bank conflicts
  const int tid  = threadIdx.x;
  const int lane = tid & 31;
  const int wave = tid >> 5;              // 0..7
  const int M    = lane & 15;             // A row / B col / D col within tile
  const int kh   = lane >> 4;             // K half-select (0: K0..1, 1: K0+2..3)

  // ---- preload B fragments + bias for this wave's NT column tiles ----
  v2f   bf[NT][32];
  float bias[NT];
#pragma unroll
  for (int j = 0; j < NT; ++j) {
    const int col0 = (wave + 8 * j) * 16;
    bias[j] = b[col0 + M];
#pragma unroll
    for (int ki = 0; ki < 32; ++ki) {
      const int k = 4 * ki + 2 * kh;
      bf[j][ki].x = W[(size_t)(k + 0) * Cout + col0 + M];
      bf[j][ki].y = W[(size_t)(k + 1) * Cout + col0 + M];
    }
  }

  const int nTilesR = (n + 15) >> 4;
  const int t0 = blockIdx.x * ROW_TILES_PER_BLOCK;
  for (int rt = 0; rt < ROW_TILES_PER_BLOCK; ++rt) {
    const int tile = t0 + rt;
    if (tile >= nTilesR) break;           // uniform across block
    const int row0 = tile * 16;

    __syncthreads();                      // protect LDS from previous readers
    for (int i = tid; i < 16 * 32; i += 256) {   // 32 float4 per row
      const int r = i >> 5, c4 = i & 31;
      const int gr = row0 + r;
      float4 v = make_float4(0.f, 0.f, 0.f, 0.f);
      if (gr < n) v = ((const float4*)(h + (size_t)gr * 128))[c4];
      hs[r][c4 * 4 + 0] = v.x; hs[r][c4 * 4 + 1] = v.y;
      hs[r][c4 * 4 + 2] = v.z; hs[r][c4 * 4 + 3] = v.w;
    }
    __syncthreads();

#pragma unroll
    for (int j = 0; j < NT; ++j) {
      v8f acc = {};
#pragma unroll
      for (int ki = 0; ki < 32; ++ki) {
        // A fragment: lane holds A[M][4*ki+2*kh .. +1] (8B-aligned ds b64)
        const v2f a = *(const v2f*)(&hs[M][4 * ki + 2 * kh]);
        acc = __builtin_amdgcn_wmma_f32_16x16x4_f32(
            /*neg_a=*/false, a, /*neg_b=*/false, bf[j][ki],
            /*c_mod=*/(short)0, acc, /*reuse_a=*/false, /*reuse_b=*/false);
      }
      const int col0 = (wave + 8 * j) * 16;
#pragma unroll
      for (int r = 0; r < 8; ++r) {       // VGPR r -> rows r (lanes<16), r+8
        const int row = row0 + r + 8 * kh;
        if (row < n) out[(size_t)row * Cout + col0 + M] = acc[r] + bias[j];
      }
    }
  }
}

// ---------------------------------------------------------------------------
// Sparse helpers
// ---------------------------------------------------------------------------
__global__ void gnn_fill(float* p, float v, int count) {
  int i = blockIdx.x * blockDim.x + threadIdx.x;
  if (i < count) p[i] = v;
}

__global__ void gnn_edge_count(const int* __restrict__ dst, int E, float* __restrict__ cnt) {
  int e = blockIdx.x * blockDim.x + threadIdx.x;
  if (e < E) atomicAdd(&cnt[dst[e]], 1.0f);
}

// wave per edge: acc[dst] += z[src]  (128 f32)
__global__ __launch_bounds__(256)
void gnn_scatter_add128(const float* __restrict__ z, const int* __restrict__ src,
                        const int* __restrict__ dst, int E, float* __restrict__ acc) {
  const int e    = (int)(((size_t)blockIdx.x * blockDim.x + threadIdx.x) >> 5);
  const int lane = threadIdx.x & 31;
  if (e >= E) return;
  // prefetch a future edge's source row (random gather -> hide L2 latency)
  const int ep = e + PF_DIST;
  if (ep < E) __builtin_prefetch(z + (size_t)src[ep] * 128 + lane * 4, 0, 0);
  const int s = src[e], d = dst[e];
  const float4 v = ((const float4*)(z + (size_t)s * 128))[lane];
  float* p = acc + (size_t)d * 128 + lane * 4;
  atomicAdd(p + 0, v.x); atomicAdd(p + 1, v.y);
  atomicAdd(p + 2, v.z); atomicAdd(p + 3, v.w);
}

// block(128) per node: out = acc/max(cnt,1), optional row L2-normalize
__global__ __launch_bounds__(128)
void gnn_seg_mean(const float* __restrict__ acc, const float* __restrict__ cnt,
                  float* __restrict__ out, int n, int donorm) {
  const int i = blockIdx.x, c = threadIdx.x;
  const float cm = fmaxf(cnt[i], 1.0f);
  float v = acc[(size_t)i * 128 + c] / cm;
  if (donorm) {
    __shared__ float red[128];
    red[c] = v * v;
    __syncthreads();
    for (int s = 64; s > 0; s >>= 1) {
      if (c < s) red[c] += red[c + s];
      __syncthreads();
    }
    v = v / fmaxf(sqrtf(red[0]), EPS_NORM);
  }
  out[(size_t)i * 128 + c] = v;
}

// wave per node: a_src = z.attW[:256], a_dst = z.attW[256:]
__global__ __launch_bounds__(256)
void gnn_att_dots(const float* __restrict__ zatt, const float* __restrict__ attW,
                  float* __restrict__ asrc, float* __restrict__ adst, int n) {
  const int node = (int)(((size_t)blockIdx.x * blockDim.x + threadIdx.x) >> 5);
  const int lane = threadIdx.x & 31;
  if (node >= n) return;
  const float* zr = zatt + (size_t)node * C1;
  float ss = 0.f, dd = 0.f;
  for (int j = lane; j < C1; j += 32) {
    const float z = zr[j];
    ss += z * attW[j];
    dd += z * attW[C1 + j];
  }
  for (int off = 16; off > 0; off >>= 1) {
    ss += __shfl_down(ss, off, 32);
    dd += __shfl_down(dd, off, 32);
  }
  if (lane == 0) { asrc[node] = ss; adst[node] = dd; }
}

__device__ __forceinline__ float gnn_leaky(float x) {
  return x > 0.f ? x : NEG_SLOPE * x;
}

__global__ void gnn_edge_max(const int* __restrict__ src, const int* __restrict__ dst,
                             const float* __restrict__ asrc, const float* __restrict__ adst,
                             const float* __restrict__ attb, float* __restrict__ mmax, int E) {
  int e = blockIdx.x * blockDim.x + threadIdx.x;
  if (e >= E) return;
  const int d = dst[e];
  const float v = gnn_leaky(asrc[src[e]] + adst[d] + attb[0]);
  // float atomic max via ordered-int trick (mmax pre-filled with -inf)
  if (v >= 0.f) atomicMax((int*)(mmax + d), __float_as_int(v));
  else          atomicMin((unsigned int*)(mmax + d), __float_as_uint(v));
}

// wave per edge: alpha = exp(e - m[dst]); denom[dst]+=alpha; num[dst]+=alpha*z[src]
__global__ __launch_bounds__(256)
void gnn_edge_scatter_att(const int* __restrict__ src, const int* __restrict__ dst,
                          const float* __restrict__ asrc, const float* __restrict__ adst,
                          const float* __restrict__ attb, const float* __restrict__ mmax,
                          const float* __restrict__ zatt, float* __restrict__ denom,
                          float* __restrict__ num, int E) {
  const int e    = (int)(((size_t)blockIdx.x * blockDim.x + threadIdx.x) >> 5);
  const int lane = threadIdx.x & 31;
  if (e >= E) return;
  const int ep = e + PF_DIST;
  if (ep < E) __builtin_prefetch(zatt + (size_t)src[ep] * C1 + lane * 8, 0, 0);
  const int s = src[e], d = dst[e];
  const float ev    = gnn_leaky(asrc[s] + adst[d] + attb[0]);
  const float alpha = __expf(ev - mmax[d]);
  if (lane == 0) atomicAdd(&denom[d], alpha);
  const float4* zr = (const float4*)(zatt + (size_t)s * C1);
  float* nr = num + (size_t)d * C1 + lane * 8;
  const float4 v0 = zr[lane * 2 + 0];
  const float4 v1 = zr[lane * 2 + 1];
  atomicAdd(nr + 0, alpha * v0.x); atomicAdd(nr + 1, alpha * v0.y);
  atomicAdd(nr + 2, alpha * v0.z); atomicAdd(nr + 3, alpha * v0.w);
  atomicAdd(nr + 4, alpha * v1.x); atomicAdd(nr + 5, alpha * v1.y);
  atomicAdd(nr + 6, alpha * v1.z); atomicAdd(nr + 7, alpha * v1.w);
}

// block(256) per node: s1 row = softmax(num/denom)
__global__ __launch_bounds__(256)
void gnn_softmax_rows(const float* __restrict__ num, const float* __restrict__ denom,
                      float* __restrict__ s1, int n) {
  const int i = blockIdx.x, c = threadIdx.x;
  const float v = num[(size_t)i * C1 + c] / denom[i];
  __shared__ float red[C1];
  red[c] = v;
  __syncthreads();
  for (int s = 128; s > 0; s >>= 1) {
    if (c < s) red[c] = fmaxf(red[c], red[c + s]);
    __syncthreads();
  }
  const float mx = red[0];
  __syncthreads();
  const float ev = __expf(v - mx);
  red[c] = ev;
  __syncthreads();
  for (int s = 128; s > 0; s >>= 1) {
    if (c < s) red[c] += red[c + s];
    __syncthreads();
  }
  s1[(size_t)i * C1 + c] = ev / red[0];
}

// x1[k,c] = sum_i s1[i,k]*z1[i,c]; split-K partials -> one atomicAdd each.
// grid = (32 k-groups of 8, SPLITS), block = 128 (c)
__global__ __launch_bounds__(128)
void gnn_xT_gemm(const float* __restrict__ s1, const float* __restrict__ z1,
                 float* __restrict__ x1, int n) {
  const int kg = blockIdx.x * 8;
  const int c  = threadIdx.x;
  const int chunk = (n + gridDim.y - 1) / gridDim.y;
  const int i0 = blockIdx.y * chunk;
  const int i1 = min(i0 + chunk, n);
  float acc[8] = {0.f, 0.f, 0.f, 0.f, 0.f, 0.f, 0.f, 0.f};
  for (int i = i0; i < i1; ++i) {
    const float zv = z1[(size_t)i * 128 + c];
    const float* sr = s1 + (size_t)i * C1 + kg;
#pragma unroll
    for (int j = 0; j < 8; ++j) acc[j] += sr[j] * zv;
  }
#pragma unroll
  for (int j = 0; j < 8; ++j) atomicAdd(&x1[(kg + j) * 128 + c], acc[j]);
}

// Coarse levels, collapsed exactly (z2 rows are a broadcast of mean(lin2)):
//   m128 = mean_rows(x1) @ We2 + be2
//   p32  = softmax(m128 @ Wa2 + ba2)          (gat_dense alpha is exactly 1/256)
//   s2 rows = p32 ; x2[k,c] = 256*p32[k]*m128[c] ; embed0 = sum(m128)/16
__global__ __launch_bounds__(256)
void gnn_tail(const float* __restrict__ x1, const float* __restrict__ We2,
              const float* __restrict__ be2, const float* __restrict__ Wa2,
              const float* __restrict__ ba2, float* __restrict__ s2,
              float* __restrict__ assign1, float* __restrict__ x2,
              float* __restrict__ embed0) {
  __shared__ float meanx[128], m128[128], p32[C2];
  const int t = threadIdx.x;
  if (t < 128) {
    float s = 0.f;
    for (int r = 0; r < C1; ++r) s += x1[r * 128 + t];
    meanx[t] = s * (1.0f / C1);
  }
  __syncthreads();
  if (t < 128) {
    float s = be2[t];
    for (int k = 0; k < 128; ++k) s += meanx[k] * We2[k * 128 + t];
    m128[t] = s;
  }
  __syncthreads();
  if (t < C2) {
    float s = ba2[t];
    for (int k = 0; k < 128; ++k) s += m128[k] * Wa2[k * C2 + t];
    p32[t] = s;
  }
  __syncthreads();
  if (t == 0) {
    float mx = p32[0];
    for (int j = 1; j < C2; ++j) mx = fmaxf(mx, p32[j]);
    float sum = 0.f;
    for (int j = 0; j < C2; ++j) { p32[j] = __expf(p32[j] - mx); sum += p32[j]; }
    for (int j = 0; j < C2; ++j) p32[j] /= sum;
  }
  __syncthreads();
  for (int idx = t; idx < C1 * C2; idx += 256) s2[idx] = p32[idx & (C2 - 1)];
  if (t < C2) assign1[t] = 1.0f;
  for (int idx = t; idx < C2 * 128; idx += 256) {
    const int k = idx >> 7, c = idx & 127;
    x2[idx] = 256.0f * p32[k] * m128[c];
  }
  if (t == 0) {
    float s = 0.f;
    for (int c = 0; c < 128; ++c) s += m128[c];
    embed0[0] = s * (1.0f / 16.0f);
  }
}

// ---------------------------------------------------------------------------
extern "C" void kernel_launch(void* const* d_in, const int* in_sizes, int n_in,
                              void* d_out, int out_size, void* d_ws, size_t ws_size,
                              hipStream_t stream) {
  const float* feature = (const float*)d_in[0];
  const int*   ei      = (const int*)d_in[1];
  const float* Wf  = (const float*)d_in[2];
  const float* bf  = (const float*)d_in[3];
  const float* We1 = (const float*)d_in[4];
  const float* be1 = (const float*)d_in[5];
  const float* Wa1 = (const float*)d_in[6];
  const float* ba1 = (const float*)d_in[7];
  const float* attW1 = (const float*)d_in[8];
  const float* attb1 = (const float*)d_in[9];
  const float* We2 = (const float*)d_in[10];
  const float* be2 = (const float*)d_in[11];
  const float* Wa2 = (const float*)d_in[12];
  const float* ba2 = (const float*)d_in[13];

  const int n = in_sizes[0] / 128;   // 20000
  const int E = in_sizes[1] / 2;     // 340000
  const int* srcp = ei;
  const int* dstp = ei + E;

  // output layout (reference tuple order, flat)
  float* out  = (float*)d_out;
  float* o_s1 = out;                                   // [n, 256]
  float* o_s2 = o_s1 + (size_t)n * C1;                 // [256, 32]
  float* o_a1 = o_s2 + (size_t)C1 * C2;                // [32, 1]
  float* o_e3 = o_a1 + C2;                             // [n, 128]  (embed3 = x)
  float* o_x1 = o_e3 + (size_t)n * 128;                // [256, 128]
  float* o_x2 = o_x1 + (size_t)C1 * 128;               // [32, 128]
  float* o_e0 = o_x2 + (size_t)C2 * 128;               // scalar

  // workspace layout
  float* ws    = (float*)d_ws;
  float* acc   = ws;                                   // n*256 (128-acc / num)
  float* gemm  = acc  + (size_t)n * C1;                // n*128 (zf / ze1)
  float* z1    = gemm + (size_t)n * 128;               // n*128
  float* zatt  = z1   + (size_t)n * 128;               // n*256
  float* cnt   = zatt + (size_t)n * C1;                // n
  float* asrc  = cnt  + n;                             // n
  float* adst  = asrc + n;                             // n
  float* mmax  = adst + n;                             // n
  float* denom = mmax + n;                             // n

  const int nTilesR = (n + 15) / 16;
  const int gBlk   = (nTilesR + ROW_TILES_PER_BLOCK - 1) / ROW_TILES_PER_BLOCK;
  const int eBlk   = (E + 255) / 256;
  const int eWaveB = (int)(((size_t)E * 32 + 255) / 256);
  const int nWaveB = (int)(((size_t)n * 32 + 255) / 256);

  // ---- GCN_f1: zf = feature@Wf+bf ; x = seg_mean ; L2-normalize -> embed3
  hipMemsetAsync(acc, 0, (size_t)n * 128 * sizeof(float), stream);
  hipMemsetAsync(cnt, 0, (size_t)n * sizeof(float), stream);
  gnn_wmma_gemm_bias<1><<<gBlk, 256, 0, stream>>>(feature, Wf, bf, gemm, n, 128);
  gnn_edge_count<<<eBlk, 256, 0, stream>>>(dstp, E, cnt);
  gnn_scatter_add128<<<eWaveB, 256, 0, stream>>>(gemm, srcp, dstp, E, acc);
  gnn_seg_mean<<<n, 128, 0, stream>>>(acc, cnt, o_e3, n, 1);

  // ---- GCN_emb (assign layer 1): z1 = seg_mean(x@We1+be1)
  gnn_wmma_gemm_bias<1><<<gBlk, 256, 0, stream>>>(o_e3, We1, be1, gemm, n, 128);
  hipMemsetAsync(acc, 0, (size_t)n * 128 * sizeof(float), stream);
  gnn_scatter_add128<<<eWaveB, 256, 0, stream>>>(gemm, srcp, dstp, E, acc);
  gnn_seg_mean<<<n, 128, 0, stream>>>(acc, cnt, z1, n, 0);

  // ---- GAT sparse: zatt = z1@Wa1+ba1 ; edge softmax ; num/denom
  gnn_wmma_gemm_bias<2><<<gBlk, 256, 0, stream>>>(z1, Wa1, ba1, zatt, n, C1);
  gnn_att_dots<<<nWaveB, 256, 0, stream>>>(zatt, attW1, asrc, adst, n);
  gnn_fill<<<(n + 255) / 256, 256, 0, stream>>>(mmax, -INFINITY, n);
  hipMemsetAsync(denom, 0, (size_t)n * sizeof(float), stream);
  hipMemsetAsync(acc, 0, (size_t)n * C1 * sizeof(float), stream);
  gnn_edge_max<<<eBlk, 256, 0, stream>>>(srcp, dstp, asrc, adst, attb1, mmax, E);
  gnn_edge_scatter_att<<<eWaveB, 256, 0, stream>>>(srcp, dstp, asrc, adst, attb1,
                                                   mmax, zatt, denom, acc, E);
  gnn_softmax_rows<<<n, 256, 0, stream>>>(acc, denom, o_s1, n);

  // ---- x1 = s1^T @ z1 (split-K + atomics)
  hipMemsetAsync(o_x1, 0, (size_t)C1 * 128 * sizeof(float), stream);
  gnn_xT_gemm<<<dim3(C1 / 8, 40), 128, 0, stream>>>(o_s1, z1, o_x1, n);

  // ---- collapsed coarse levels: s2, assign1, x2, embed0
  gnn_tail<<<1, 256, 0, stream>>>(o_x1, We2, be2, Wa2, ba2, o_s2, o_a1, o_x2, o_e0);
}